// CustomLoss_70257075028730
// MI455X (gfx1250) — compile-verified
//
#include <hip/hip_runtime.h>

// ---------------- problem constants (SSD300/VOC loss) ----------------
#define B_    64
#define N_    8732
#define C_    21
#define BN_   (B_ * N_)            // 558848 = 2183 * 256 (exact)
#define TB_   256                  // boxes per block in the box kernel
#define NBLK_ (BN_ / TB_)          // 2183
#define TILE_FLOATS (TB_ * C_)     // 5376 floats per staged label tile
#define TILE_BYTES  (TILE_FLOATS * 4)   // 21504 bytes (16B aligned per tile)
#define TILE_CHUNKS (TILE_BYTES / 16)   // 1344 x b128 async chunks

typedef __attribute__((ext_vector_type(2))) float v2f;
typedef __attribute__((ext_vector_type(8))) float v8f;

// ---------------- CDNA5 async global->LDS path (guarded) ----------------
// Builtin params are pointers to a 16-byte int vector (per hipcc diagnostic):
//   param0: __attribute__((vector_size(16))) int  addrspace(1)*
//   param1: __attribute__((vector_size(16))) int  addrspace(3)*
typedef int v4i_ __attribute__((vector_size(16)));

#if __has_builtin(__builtin_amdgcn_global_load_async_to_lds_b128)
#define HAVE_ASYNC_LDS 1
#define ASG(p) ((__attribute__((address_space(1))) v4i_*)(p))
#define ASL(p) ((__attribute__((address_space(3))) v4i_*)(p))
#else
#warning "no __builtin_amdgcn_global_load_async_to_lds_b128; using VALU staging fallback"
#define HAVE_ASYNC_LDS 0
#endif

__device__ __forceinline__ void wait_async_zero() {
#if __has_builtin(__builtin_amdgcn_s_wait_asynccnt)
  __builtin_amdgcn_s_wait_asynccnt(0);
#else
  asm volatile("s_wait_asynccnt 0" ::: "memory");
#endif
}

// D = A(16x4) * B(4x16) + C, f32 WMMA (CDNA5)
#define WMMA4(Av, Bv, Cv) \
  __builtin_amdgcn_wmma_f32_16x16x4_f32(false, (Av), false, (Bv), (short)0, (Cv), false, false)

__device__ __forceinline__ float huber1(float d) {
  float a = fabsf(d);
  return (a <= 1.0f) ? 0.5f * d * d : (a - 0.5f);
}

// =====================================================================
// Kernel 1: per-box pass. Stages label tiles into LDS with async b128
// copies, computes pos/huber (loc loss), and per-box CE-from-logits
// (ranking key) + CE-from-probs via chained V_WMMA_F32_16X16X4_F32
// class-dim reductions. Writes ce_rank / ce_probs rows for the miner.
// =====================================================================
__global__ __launch_bounds__(256) void ssd_box_kernel(
    const float* __restrict__ a_del,   // [B,N,4]
    const float* __restrict__ a_lab,   // [B,N,C]
    const float* __restrict__ p_del,   // [B,N,4]
    const float* __restrict__ p_lab,   // [B,N,C]
    float* __restrict__ ce_rank,       // [B*N] masked ranking CE (-inf on pos)
    float* __restrict__ ce_probs,      // [B*N]
    int*   __restrict__ pos_count,     // [B]
    float* __restrict__ loc_sum,       // [1]
    float* __restrict__ conf_sum)      // [1] (positives contribution here)
{
  __shared__ __attribute__((aligned(16))) float sY[TILE_FLOATS]; // actual_labels tile
  __shared__ __attribute__((aligned(16))) float sX[TILE_FLOATS]; // pred_labels tile
  __shared__ float sD[8][16][17];   // per-wave 16x16 D-matrix dump (padded)
  __shared__ float sMx[8][16];      // per-wave per-box max logit
  __shared__ float sSx[8][16];      // per-wave per-box sum(x)
  __shared__ float sSe[8][16];      // per-wave per-box sum(exp(x-mx))
  __shared__ float sPos[TB_];       // per-box positive flag
  __shared__ float sLoc;            // block loc-loss partial

  const int tid  = threadIdx.x;
  const int blk  = blockIdx.x;
  const int wave = tid >> 5;
  const int lane = tid & 31;

  // ---- stage both label tiles into LDS ----
#if HAVE_ASYNC_LDS
  {
    const char* gy = (const char*)a_lab + (size_t)blk * TILE_BYTES;
    const char* gx = (const char*)p_lab + (size_t)blk * TILE_BYTES;
    for (int i = tid; i < TILE_CHUNKS; i += 256) {
      __builtin_amdgcn_global_load_async_to_lds_b128(
          ASG(gy + (size_t)i * 16), ASL((char*)sY + (size_t)i * 16), 0, 0);
      __builtin_amdgcn_global_load_async_to_lds_b128(
          ASG(gx + (size_t)i * 16), ASL((char*)sX + (size_t)i * 16), 0, 0);
    }
    wait_async_zero();
  }
#else
  {
    const float* gy = a_lab + (size_t)blk * TILE_FLOATS;
    const float* gx = p_lab + (size_t)blk * TILE_FLOATS;
    for (int i = tid; i < TILE_FLOATS; i += 256) { sY[i] = gy[i]; sX[i] = gx[i]; }
  }
#endif
  if (tid == 0) sLoc = 0.0f;
  __syncthreads();

  // ---- per-thread box: pos flag + Huber loc loss ----
  {
    const size_t gbox = (size_t)blk * TB_ + tid;
    const float4 ad = ((const float4*)a_del)[gbox];
    const float4 pd = ((const float4*)p_del)[gbox];
    const bool pos = (ad.x != 0.f) | (ad.y != 0.f) | (ad.z != 0.f) | (ad.w != 0.f);
    float h = huber1(pd.x - ad.x) + huber1(pd.y - ad.y) +
              huber1(pd.z - ad.z) + huber1(pd.w - ad.w);
    h *= 0.25f;                                   // mean over last axis
    sPos[tid] = pos ? 1.0f : 0.0f;
    if (pos) {
      atomicAdd(&sLoc, h);
      atomicAdd(&pos_count[(int)(gbox / N_)], 1);
    }
  }
  __syncthreads();
  if (tid == 0) atomicAdd(loc_sum, sLoc);

  // ---- per-wave WMMA pass: 2 tiles of 16 boxes each ----
  const int moff = (lane >= 16) ? 2 : 0;          // K sub-offset per WMMA A/B layout
  const int mrow = lane & 15;                     // box within tile (M for A, N for B)

  for (int tt = 0; tt < 2; ++tt) {
    const int tb = wave * 32 + tt * 16;           // block-local base box of tile
    const int bm = tb + mrow;                     // this lane's box for A/B elements

    // phase A: per-box max logit (lanes 0-15)
    if (lane < 16) {
      float m = -3.4e38f;
      const float* xr = &sX[(tb + lane) * C_];
      for (int c = 0; c < C_; ++c) m = fmaxf(m, xr[c]);
      sMx[wave][lane] = m;
    }
    __syncthreads();
    const float mxv = sMx[wave][mrow];

    // phase B: chained 16x16x4 f32 WMMAs over the class dim (padded to 24)
    v8f c_sx  = {0,0,0,0,0,0,0,0};   // row-sums of x       (B = ones splat)
    v8f c_se  = {0,0,0,0,0,0,0,0};   // row-sums of exp     (B = ones splat)
    v8f c_dyx = {0,0,0,0,0,0,0,0};   // Y * X^T  (diag = y.x per box)
    const v2f vone = {1.0f, 1.0f};
#pragma unroll
    for (int kk = 0; kk < 6; ++kk) {
      const int k = kk * 4 + moff;
      const float x0 = (k     < C_) ? sX[bm * C_ + k]     : 0.0f;
      const float x1 = (k + 1 < C_) ? sX[bm * C_ + k + 1] : 0.0f;
      const float y0 = (k     < C_) ? sY[bm * C_ + k]     : 0.0f;
      const float y1 = (k + 1 < C_) ? sY[bm * C_ + k + 1] : 0.0f;
      const float e0 = (k     < C_) ? __expf(x0 - mxv)    : 0.0f;
      const float e1 = (k + 1 < C_) ? __expf(x1 - mxv)    : 0.0f;
      const v2f ax = {x0, x1};
      const v2f ay = {y0, y1};
      const v2f ae = {e0, e1};
      c_sx  = WMMA4(ax, vone, c_sx);
      c_se  = WMMA4(ae, vone, c_se);
      c_dyx = WMMA4(ay, ax,  c_dyx);
    }
    // splat-B results: every column equal; lane0 holds M=0..7, lane16 M=8..15
    if (lane == 0 || lane == 16) {
      const int b0 = (lane == 0) ? 0 : 8;
#pragma unroll
      for (int v = 0; v < 8; ++v) { sSx[wave][b0 + v] = c_sx[v]; sSe[wave][b0 + v] = c_se[v]; }
    }
    __syncthreads();

    // phase C: Y * log(clip(X/sum))^T  (diag = CE-from-probs numerator)
    const float inv_sx = 1.0f / sSx[wave][mrow];
    v8f c_du = {0,0,0,0,0,0,0,0};
#pragma unroll
    for (int kk = 0; kk < 6; ++kk) {
      const int k = kk * 4 + moff;
      const float x0 = (k     < C_) ? sX[bm * C_ + k]     : 0.0f;
      const float x1 = (k + 1 < C_) ? sX[bm * C_ + k + 1] : 0.0f;
      const float y0 = (k     < C_) ? sY[bm * C_ + k]     : 0.0f;
      const float y1 = (k + 1 < C_) ? sY[bm * C_ + k + 1] : 0.0f;
      const float p0 = fminf(fmaxf(x0 * inv_sx, 1e-7f), 1.0f - 1e-7f);
      const float p1 = fminf(fmaxf(x1 * inv_sx, 1e-7f), 1.0f - 1e-7f);
      const v2f ay = {y0, y1};
      const v2f bu = {__logf(p0), __logf(p1)};
      c_du = WMMA4(ay, bu, c_du);
    }

    // dump D matrices to LDS and read the diagonals
    const int mb = (lane < 16) ? 0 : 8;
    const int nc = lane & 15;
#pragma unroll
    for (int v = 0; v < 8; ++v) sD[wave][mb + v][nc] = c_dyx[v];
    __syncthreads();
    const float dyx = (lane < 16) ? sD[wave][lane][lane] : 0.0f;
    __syncthreads();
#pragma unroll
    for (int v = 0; v < 8; ++v) sD[wave][mb + v][nc] = c_du[v];
    __syncthreads();
    const float dup = (lane < 16) ? sD[wave][lane][lane] : 0.0f;

    // finalize 16 boxes (lanes 0-15)
    if (lane < 16) {
      const int   lbox = tb + lane;
      const size_t gbox = (size_t)blk * TB_ + lbox;
      const bool  pos  = sPos[lbox] != 0.0f;
      const float lse  = sMx[wave][lane] + __logf(sSe[wave][lane]); // logsumexp
      const float cr   = pos ? -__builtin_inff() : (lse - dyx);     // ranking CE
      const float cp   = -dup;                                      // CE from probs
      ce_rank[gbox]  = cr;
      ce_probs[gbox] = cp;
      if (pos) atomicAdd(conf_sum, cp);                             // positives part
    }
    __syncthreads();
  }
}

// =====================================================================
// Kernel 2: hard-negative mining. One block per batch row; row keys
// live in LDS (35KB << 320KB/WGP); 32-step radix bisection finds the
// k-th largest CE among negatives; sum ce_probs of selected negatives.
// =====================================================================
__global__ __launch_bounds__(256) void ssd_mine_kernel(
    const float* __restrict__ ce_rank,
    const float* __restrict__ ce_probs,
    const int*   __restrict__ pos_count,
    float* __restrict__ conf_sum)
{
  __shared__ unsigned sKey[N_];
  __shared__ int   sCnt;
  __shared__ float sSum;

  const int tid = threadIdx.x;
  const int b   = blockIdx.x;

  // order-preserving float->uint key; -inf (positives) maps very low
  for (int i = tid; i < N_; i += 256) {
    unsigned u = __float_as_uint(ce_rank[(size_t)b * N_ + i]);
    sKey[i] = (u & 0x80000000u) ? ~u : (u | 0x80000000u);
  }
  __syncthreads();

  const int k = 3 * pos_count[b];   // NEG_POS_RATIO * positives

  unsigned T = 0u;
  for (int bit = 31; bit >= 0; --bit) {
    const unsigned cand = T | (1u << bit);
    __syncthreads();
    if (tid == 0) sCnt = 0;
    __syncthreads();
    int local = 0;
    for (int i = tid; i < N_; i += 256) local += (sKey[i] >= cand) ? 1 : 0;
    atomicAdd(&sCnt, local);
    __syncthreads();
    if (sCnt >= k) T = cand;        // uniform decision (sCnt is shared)
  }

  __syncthreads();
  if (tid == 0) sSum = 0.0f;
  __syncthreads();
  float local = 0.0f;
  if (k > 0) {
    for (int i = tid; i < N_; i += 256)
      if (sKey[i] >= T) local += ce_probs[(size_t)b * N_ + i];
  }
  atomicAdd(&sSum, local);
  __syncthreads();
  if (tid == 0) atomicAdd(conf_sum, sSum);
}

// =====================================================================
// Kernel 3: finalize the two scalars.
// =====================================================================
__global__ void ssd_finalize_kernel(const int* __restrict__ pos_count,
                                    const float* __restrict__ loc_sum,
                                    const float* __restrict__ conf_sum,
                                    float* __restrict__ out)
{
  if (threadIdx.x == 0 && blockIdx.x == 0) {
    int s = 0;
    for (int i = 0; i < B_; ++i) s += pos_count[i];
    const float tp = fmaxf((float)s, 1.0f);
    out[0] = loc_sum[0]  / tp;   // LOC_LOSS_ALPHA == 1
    out[1] = conf_sum[0] / tp;
  }
}

// =====================================================================
// Host launcher (graph-capture safe: only async ops on `stream`)
// =====================================================================
extern "C" void kernel_launch(void* const* d_in, const int* in_sizes, int n_in,
                              void* d_out, int out_size, void* d_ws, size_t ws_size,
                              hipStream_t stream) {
  (void)in_sizes; (void)n_in; (void)out_size; (void)ws_size;
  const float* a_del = (const float*)d_in[0];   // actual_bbox_deltas [B,N,4]
  const float* a_lab = (const float*)d_in[1];   // actual_labels     [B,N,C]
  const float* p_del = (const float*)d_in[2];   // pred_bbox_deltas  [B,N,4]
  const float* p_lab = (const float*)d_in[3];   // pred_labels       [B,N,C]

  char*  ws        = (char*)d_ws;
  int*   pos_count = (int*)ws;                       // 64 ints
  float* loc_sum   = (float*)(ws + 256);             // 1 float
  float* conf_sum  = (float*)(ws + 260);             // 1 float
  float* ce_rank   = (float*)(ws + 512);             // BN floats
  float* ce_probs  = (float*)(ws + 512 + (size_t)BN_ * 4);

  (void)hipMemsetAsync(ws, 0, 512, stream);

  ssd_box_kernel<<<NBLK_, 256, 0, stream>>>(a_del, a_lab, p_del, p_lab,
                                            ce_rank, ce_probs,
                                            pos_count, loc_sum, conf_sum);
  ssd_mine_kernel<<<B_, 256, 0, stream>>>(ce_rank, ce_probs, pos_count, conf_sum);
  ssd_finalize_kernel<<<1, 1, 0, stream>>>(pos_count, loc_sum, conf_sum, (float*)d_out);
}